// DWT_Layer_14448269984447
// MI455X (gfx1250) — compile-verified
//
#include <hip/hip_runtime.h>

typedef float v2f __attribute__((ext_vector_type(2)));
typedef float v8f __attribute__((ext_vector_type(8)));

#define HW_FLAT 1048576

__device__ __forceinline__ int reflect_idx(int t, int n) {
    t = (t < 0) ? (-1 - t) : t;
    t = (t >= n) ? (2 * n - 1 - t) : t;
    return t;
}

// db4 decomposition low-pass taps (pywt order), f32
__device__ __forceinline__ float tapLo(int m) {
    float r = 0.0f;
    r = (m == 0) ? -1.0597401784997278e-02f : r;
    r = (m == 1) ?  3.2883011666982945e-02f : r;
    r = (m == 2) ?  3.0841381835986965e-02f : r;
    r = (m == 3) ? -1.8703481171888114e-01f : r;
    r = (m == 4) ? -2.7983769416983850e-02f : r;
    r = (m == 5) ?  6.3088076792959036e-01f : r;
    r = (m == 6) ?  7.1484657055254153e-01f : r;
    r = (m == 7) ?  2.3037781330885523e-01f : r;
    return r;
}
// hi[m] = (-1)^(m+1) * lo[7-m]
__device__ __forceinline__ float tapHi(int m) {
    float s = (m & 1) ? 1.0f : -1.0f;
    return s * tapLo(7 - m);
}

// One fused DWT level. Each 32-thread block (1 wave) produces one 16x16 tile of
// each of the four subbands (aa, da, ad, dd) via WMMA f32 16x16x4 banded matmuls.
//   out[j] = sum_m f[m] * xext(2j + 2 - m)   (pywt symmetric, len J = (N+7)/2)
// Tile coords: extended index t = 2*tile0 - 5 + s, s in [0,38); W[s,j] = f[2j+7-s].
__global__ __launch_bounds__(32) void dwt_level_kernel(
    const float* __restrict__ in, int N, int J,
    float* __restrict__ aaBuf,   // non-null: aa -> aaBuf (stride J*J/img); null: aa packed
    float* __restrict__ out,     // packed output base (d_out)
    int offAa, int offDa, int offAd, int offDd)
{
    __shared__ float in_tile[48 * 42];  // rows 38..47 deliberately uninit (discarded)
    __shared__ float loT[16 * 42];      // loT[j][s] : transposed row-stage result
    __shared__ float hiT[16 * 42];

    const int lane   = threadIdx.x;
    const int j16    = lane & 15;
    const int hiHalf = (lane >= 16) ? 1 : 0;
    const int kx     = hiHalf * 2;            // WMMA K-slot base for this half-wave
    const int i0     = blockIdx.y * 16;
    const int j0     = blockIdx.x * 16;
    const int img    = blockIdx.z;

    const float* src = in + (size_t)img * N * N;

    // --- load 38x38 extended input tile (cols 38..41 zero: W==0 there, avoid 0*NaN) ---
    for (int l = lane; l < 38 * 42; l += 32) {
        int sr = l / 42, sc = l - sr * 42;
        float v = 0.0f;
        if (sc < 38) {
            int gr = reflect_idx(2 * i0 - 5 + sr, N);
            int gc = reflect_idx(2 * j0 - 5 + sc, N);
            v = src[(size_t)gr * N + gc];
        }
        in_tile[l] = v;
    }
    if (lane < 16) {
        #pragma unroll
        for (int c = 38; c < 42; ++c) { loT[lane * 42 + c] = 0.0f; hiT[lane * 42 + c] = 0.0f; }
    }

    // --- per-lane filter fragments: f[2*(lane&15)+7 - (4c+kx+{0,1})] ---
    // Same values serve as row-stage B (4x16) and col-stage A (16x4) fragments.
    float flox[10], floy[10], fhix[10], fhiy[10];
    #pragma unroll
    for (int c = 0; c < 10; ++c) {
        int m0 = 2 * j16 + 7 - (4 * c + kx);
        flox[c] = tapLo(m0);     floy[c] = tapLo(m0 - 1);
        fhix[c] = tapHi(m0);     fhiy[c] = tapHi(m0 - 1);
    }

    // --- stage 1: horizontal DWT over 38 tile rows (3 groups of 16) ---
    #pragma unroll
    for (int g = 0; g < 3; ++g) {
        v8f accL = {}; v8f accH = {};
        #pragma unroll
        for (int c = 0; c < 10; ++c) {
            // A: lane holds in_tile[row = 16g + lane&15][s = 4c + kx .. +1]
            const v2f a = *reinterpret_cast<const v2f*>(
                &in_tile[(16 * g + j16) * 42 + 4 * c + kx]);
            v2f bL = {flox[c], floy[c]};
            v2f bH = {fhix[c], fhiy[c]};
            accL = __builtin_amdgcn_wmma_f32_16x16x4_f32(false, a, false, bL,
                                                         (short)0, accL, false, false);
            accH = __builtin_amdgcn_wmma_f32_16x16x4_f32(false, a, false, bH,
                                                         (short)0, accH, false, false);
        }
        // D[m][n] = lo[tile row 16g+m][out col j0+n]; store transposed: loT[n][16g+m]
        #pragma unroll
        for (int v = 0; v < 8; ++v) {
            int s = 16 * g + v + 8 * hiHalf;
            if (s < 38) {
                loT[j16 * 42 + s] = accL[v];
                hiT[j16 * 42 + s] = accH[v];
            }
        }
    }

    // --- stage 2: vertical DWT -> aa, da, ad, dd (A = filter band, B = loT/hiT) ---
    v8f aa = {}, da = {}, ad = {}, dd = {};
    #pragma unroll
    for (int c = 0; c < 10; ++c) {
        v2f aL = {flox[c], floy[c]};
        v2f aH = {fhix[c], fhiy[c]};
        const v2f bLo = *reinterpret_cast<const v2f*>(&loT[j16 * 42 + 4 * c + kx]);
        const v2f bHi = *reinterpret_cast<const v2f*>(&hiT[j16 * 42 + 4 * c + kx]);
        aa = __builtin_amdgcn_wmma_f32_16x16x4_f32(false, aL, false, bLo, (short)0, aa, false, false);
        da = __builtin_amdgcn_wmma_f32_16x16x4_f32(false, aH, false, bLo, (short)0, da, false, false);
        ad = __builtin_amdgcn_wmma_f32_16x16x4_f32(false, aL, false, bHi, (short)0, ad, false, false);
        dd = __builtin_amdgcn_wmma_f32_16x16x4_f32(false, aH, false, bHi, (short)0, dd, false, false);
    }

    // --- stores: D vgpr v -> (row i0+v+8*hiHalf, col j0+lane&15), with edge+trunc masks ---
    const int jj = j0 + j16;
    const size_t outImg = (size_t)img * HW_FLAT;
    #pragma unroll
    for (int v = 0; v < 8; ++v) {
        int ii = i0 + v + 8 * hiHalf;
        if (ii < J && jj < J) {
            int idx = ii * J + jj;
            if (aaBuf) {
                aaBuf[(size_t)img * J * J + idx] = aa[v];
            } else {
                int f0 = offAa + idx; if (f0 < HW_FLAT) out[outImg + f0] = aa[v];
            }
            int f1 = offDa + idx; if (f1 < HW_FLAT) out[outImg + f1] = da[v];
            int f2 = offAd + idx; if (f2 < HW_FLAT) out[outImg + f2] = ad[v];
            int f3 = offDd + idx; if (f3 < HW_FLAT) out[outImg + f3] = dd[v];
        }
    }
}

extern "C" void kernel_launch(void* const* d_in, const int* in_sizes, int n_in,
                              void* d_out, int out_size, void* d_ws, size_t ws_size,
                              hipStream_t stream) {
    const float* x = (const float*)d_in[0];   // (16,3,1024,1024) f32
    float* out = (float*)d_out;               // (16,3,64,128,128) f32
    // Workspace: aa level-1 (48*515*515 f32 = ~51MB) + aa level-2 (48*261*261 = ~13MB)
    float* A1 = (float*)d_ws;
    float* A2 = A1 + (size_t)48 * 515 * 515;

    dim3 blk(32, 1, 1);
    // Level 1: 1024 -> 515   (da@276187, ad@541412, dd@806637 [truncated])
    dwt_level_kernel<<<dim3(33, 33, 48), blk, 0, stream>>>(
        x, 1024, 515, A1, out, 0, 276187, 541412, 806637);
    // Level 2: 515 -> 261    (da@71824, ad@139945, dd@208066)
    dwt_level_kernel<<<dim3(17, 17, 48), blk, 0, stream>>>(
        A1, 515, 261, A2, out, 0, 71824, 139945, 208066);
    // Level 3: 261 -> 134    (aa@0, da@17956, ad@35912, dd@53868) -> packed directly
    dwt_level_kernel<<<dim3(9, 9, 48), blk, 0, stream>>>(
        A2, 261, 134, nullptr, out, 0, 17956, 35912, 53868);
}